// GinEncoder_11605001633947
// MI455X (gfx1250) — compile-verified
//
#include <hip/hip_runtime.h>
#include <hip/hip_bf16.h>
#include <math.h>

#define HD 128           // feature width everywhere (F_IN == H == 128)
#define NGRAPH_COLS 256  // output row stride: concat([p0, p1], axis=1)
#define BN_EPS 1e-5f

typedef float v2f __attribute__((ext_vector_type(2)));
typedef float v8f __attribute__((ext_vector_type(8)));

// ---------------------------------------------------------------- zero fill
__global__ void gin_zero4_kernel(float4* __restrict__ p, int n4) {
    int i = blockIdx.x * blockDim.x + threadIdx.x;
    if (i < n4) p[i] = make_float4(0.f, 0.f, 0.f, 0.f);
}

// ------------------------------------------------------- edge scatter (add)
// agg[dst] += h[src]; 32 threads per edge, float4 per thread, f32 atomics
// (feature matrix is 25.6 MB -> fully L2-resident on the 192 MB L2)
__global__ void gin_scatter_kernel(const float* __restrict__ h,
                                   const int* __restrict__ ei,
                                   float* __restrict__ agg, int nedges) {
    long long gid = (long long)blockIdx.x * blockDim.x + threadIdx.x;
    int e = (int)(gid >> 5);
    if (e >= nedges) return;
    int f = ((int)gid & 31) * 4;
    int s = ei[e];            // edge_index[0][e]
    int d = ei[nedges + e];   // edge_index[1][e]
    const float4 v = *(const float4*)(&h[(long long)s * HD + f]);
    float* ap = &agg[(long long)d * HD + f];
    atomicAdd(ap + 0, v.x);
    atomicAdd(ap + 1, v.y);
    atomicAdd(ap + 2, v.z);
    atomicAdd(ap + 3, v.w);
}

// ---------------------------------------------------------------- WMMA GEMM
// Z = f(A) @ W + bias, plus per-column sum / sum-of-squares for the BN that
// follows. MODE 0: f(A) = A0 + A1 (input + aggregation)
//          MODE 1: f(A) = relu(A0 * bn_a + bn_c) (fused BatchNorm+ReLU)
// Block: 256 thr = 8 waves; wave w -> rows [blk*128 + w*16, +16), all 128 cols.
// V_WMMA_F32_16X16X4_F32: A 16x4 (v2f/lane), B 4x16 (v2f/lane), C/D v8f.
//
// B is staged quad-interleaved in LDS: W[k][n] -> sW[(k>>2)*512 + n*4 + (k&3)]
// so each lane's fragment {B[ka][col], B[ka+1][col]} (ka = kb + 2*hi) is one
// aligned ds_load_b64 at byte offset (kb>>2)*2048 + n*256 from a fixed lane
// base (col*16 + hi*8). All offsets < 64K -> pure immediate-offset DS loads,
// and the two half-waves hit disjoint LDS banks (4c..4c+1 vs 4c+2..4c+3).
template <int MODE>
__global__ void __launch_bounds__(256)
gin_gemm_kernel(const float* __restrict__ A0, const float* __restrict__ A1,
                const float* __restrict__ bn_a, const float* __restrict__ bn_c,
                const float* __restrict__ W, const float* __restrict__ bias,
                float* __restrict__ Z, float* __restrict__ colsum,
                float* __restrict__ colsumsq, int nrows) {
    __shared__ float sW[HD * HD];  // 64 KB of the 320 KB WGP LDS
    const int tid = threadIdx.x;
    for (int i = tid * 4; i < HD * HD; i += 256 * 4) {
        const float4 w = *(const float4*)(&W[i]);
        const int k = i >> 7;
        const int n = i & 127;
        float* dq = &sW[(k >> 2) * 512 + n * 4 + (k & 3)];
        dq[0] = w.x;
        dq[4] = w.y;
        dq[8] = w.z;
        dq[12] = w.w;
    }
    __syncthreads();

    const int lane = tid & 31;
    const int wave = tid >> 5;
    const int lr = lane & 15;   // sub-tile row (A) / col-within-tile (B,D)
    const int hi = lane >> 4;   // half-wave select
    const int row0 = (blockIdx.x * 8 + wave) * 16;
    const int arow = row0 + lr;
    const float rmask = (arow < nrows) ? 1.f : 0.f;  // branch-free tail mask
    const int arc = (arow < nrows) ? arow : (nrows - 1);
    const float* __restrict__ a0p = A0 + (size_t)arc * HD;
    const float* __restrict__ a1p = (MODE == 0) ? A1 + (size_t)arc * HD : A0;
    const float* __restrict__ sB = sW + lr * 4 + hi * 2;  // lane base

    v8f acc[8];
#pragma unroll
    for (int n = 0; n < 8; ++n) acc[n] = v8f{};

#pragma unroll
    for (int kb = 0; kb < HD; kb += 4) {
        const int ka = kb + 2 * hi;  // ISA A-layout: VGPR0=K{0|2}, VGPR1=K{1|3}
        v2f av = *(const v2f*)(&a0p[ka]);  // global_load_b64
        if (MODE == 0) {
            av += *(const v2f*)(&a1p[ka]);
        } else {
            const v2f ba = *(const v2f*)(&bn_a[ka]);
            const v2f bc = *(const v2f*)(&bn_c[ka]);
            av = av * ba + bc;
            av.x = fmaxf(av.x, 0.f);
            av.y = fmaxf(av.y, 0.f);
        }
        const v2f a = av * rmask;
        const float* __restrict__ bq = sB + (kb >> 2) * 512;
#pragma unroll
        for (int n = 0; n < 8; ++n) {
            const v2f b = *(const v2f*)(bq + n * 64);  // ds_load_b64, imm offs
            acc[n] = __builtin_amdgcn_wmma_f32_16x16x4_f32(
                false, a, false, b, (short)0, acc[n], false, false);
        }
    }

    // epilogue: bias, store Z, per-column BN statistics (f32 atomics)
    if (row0 + 16 <= nrows) {  // wave-uniform fast path (all but tail block)
#pragma unroll
        for (int n = 0; n < 8; ++n) {
            const int col = n * 16 + lr;
            const float bc = bias[col];
            float s = 0.f, s2 = 0.f;
            float* zp = Z + (size_t)(row0 + 8 * hi) * HD + col;
#pragma unroll
            for (int v = 0; v < 8; ++v) {  // D layout: lanes>=16 hold M+8
                float d = acc[n][v] + bc;
                zp[v * HD] = d;
                s += d;
                s2 += d * d;
            }
            atomicAdd(&colsum[col], s);
            atomicAdd(&colsumsq[col], s2);
        }
    } else {
#pragma unroll
        for (int n = 0; n < 8; ++n) {
            const int col = n * 16 + lr;
            const float bc = bias[col];
            float s = 0.f, s2 = 0.f;
#pragma unroll
            for (int v = 0; v < 8; ++v) {
                const int r = row0 + v + 8 * hi;
                if (r < nrows) {
                    float d = acc[n][v] + bc;
                    Z[(size_t)r * HD + col] = d;
                    s += d;
                    s2 += d * d;
                }
            }
            atomicAdd(&colsum[col], s);
            atomicAdd(&colsumsq[col], s2);
        }
    }
}

// ------------------------------------------------- fold BN stats into a*z+c
__global__ void gin_stats_kernel(const float* __restrict__ sum,
                                 const float* __restrict__ sumsq,
                                 const float* __restrict__ g,
                                 const float* __restrict__ be,
                                 float* __restrict__ bn_a,
                                 float* __restrict__ bn_c, float invN) {
    int c = threadIdx.x;  // 128 threads
    float m = sum[c] * invN;
    float var = fmaf(-m, m, sumsq[c] * invN);  // biased variance (jnp.var)
    float a = g[c] * rsqrtf(var + BN_EPS);
    bn_a[c] = a;
    bn_c[c] = fmaf(-m, a, be[c]);
}

// -------------------------------- BN+ReLU epilogue + global add-pool output
__global__ void gin_finish_kernel(const float* __restrict__ Z,
                                  const float* __restrict__ bn_a,
                                  const float* __restrict__ bn_c,
                                  const int* __restrict__ batch,
                                  float* __restrict__ hout,  // may be null
                                  float* __restrict__ pool, int pool_off,
                                  int nrows) {
    long long gid = (long long)blockIdx.x * blockDim.x + threadIdx.x;
    int r = (int)(gid >> 5);
    if (r >= nrows) return;
    int c = ((int)gid & 31) * 4;
    const float4 z = *(const float4*)(&Z[(long long)r * HD + c]);
    float d0 = fmaxf(fmaf(z.x, bn_a[c + 0], bn_c[c + 0]), 0.f);
    float d1 = fmaxf(fmaf(z.y, bn_a[c + 1], bn_c[c + 1]), 0.f);
    float d2 = fmaxf(fmaf(z.z, bn_a[c + 2], bn_c[c + 2]), 0.f);
    float d3 = fmaxf(fmaf(z.w, bn_a[c + 3], bn_c[c + 3]), 0.f);
    if (hout) {
        float4 hv = make_float4(d0, d1, d2, d3);
        *(float4*)(&hout[(long long)r * HD + c]) = hv;
    }
    float* pp = &pool[batch[r] * NGRAPH_COLS + pool_off + c];
    atomicAdd(pp + 0, d0);
    atomicAdd(pp + 1, d1);
    atomicAdd(pp + 2, d2);
    atomicAdd(pp + 3, d3);
}

// --------------------------------------------------------------------------
static inline void zero4(float* p, int nfloats, hipStream_t s) {
    int n4 = nfloats / 4;  // all our buffers are multiples of 4 floats
    gin_zero4_kernel<<<(n4 + 255) / 256, 256, 0, s>>>((float4*)p, n4);
}

extern "C" void kernel_launch(void* const* d_in, const int* in_sizes, int n_in,
                              void* d_out, int out_size, void* d_ws,
                              size_t ws_size, hipStream_t stream) {
    const float* x     = (const float*)d_in[0];
    const int*   ei    = (const int*)d_in[1];
    const int*   batch = (const int*)d_in[2];
    const float* Wp[4] = {(const float*)d_in[3], (const float*)d_in[7],
                          (const float*)d_in[11], (const float*)d_in[15]};
    const float* bp[4] = {(const float*)d_in[4], (const float*)d_in[8],
                          (const float*)d_in[12], (const float*)d_in[16]};
    const float* gp[4] = {(const float*)d_in[5], (const float*)d_in[9],
                          (const float*)d_in[13], (const float*)d_in[17]};
    const float* ep[4] = {(const float*)d_in[6], (const float*)d_in[10],
                          (const float*)d_in[14], (const float*)d_in[18]};

    const int nnodes = in_sizes[0] / HD;
    const int nedges = in_sizes[1] / 2;
    const float invN = 1.f / (float)nnodes;

    const size_t NF = (size_t)nnodes * HD;
    float* ws   = (float*)d_ws;
    float* agg  = ws;            // also reused as z2 (dead during GEMM2)
    float* z1   = ws + NF;
    float* h0   = ws + 2 * NF;
    float* csum = ws + 3 * NF;   // [128] col sums
    float* csq  = csum + HD;     // [128] col sum-of-squares
    float* bna  = csq + HD;      // [128] folded BN scale
    float* bnc  = bna + HD;      // [128] folded BN shift
    float* out  = (float*)d_out;

    const int gGemm    = (nnodes + 127) / 128;
    const int gScatter = (nedges * 32 + 255) / 256;
    const int gElem    = (nnodes * 32 + 255) / 256;

    zero4(out, out_size, stream);

    for (int layer = 0; layer < 2; ++layer) {
        const float* hin = (layer == 0) ? x : h0;
        const int pi = 2 * layer;  // param index of first linear

        zero4(agg, (int)NF, stream);
        zero4(csum, 2 * HD, stream);
        gin_scatter_kernel<<<gScatter, 256, 0, stream>>>(hin, ei, agg, nedges);

        // z1 = (hin + agg) @ W1 + b1, with column stats
        gin_gemm_kernel<0><<<gGemm, 256, 0, stream>>>(
            hin, agg, nullptr, nullptr, Wp[pi], bp[pi], z1, csum, csq, nnodes);
        gin_stats_kernel<<<1, HD, 0, stream>>>(csum, csq, gp[pi], ep[pi], bna,
                                               bnc, invN);

        // z2 = relu(bn(z1)) @ W2 + b2 (z2 stored in agg buffer), with stats
        zero4(csum, 2 * HD, stream);
        gin_gemm_kernel<1><<<gGemm, 256, 0, stream>>>(
            z1, nullptr, bna, bnc, Wp[pi + 1], bp[pi + 1], agg, csum, csq,
            nnodes);
        gin_stats_kernel<<<1, HD, 0, stream>>>(csum, csq, gp[pi + 1],
                                               ep[pi + 1], bna, bnc, invN);

        // h = relu(bn(z2)); store h0 for next layer (layer 0 only); pool
        gin_finish_kernel<<<gElem, 256, 0, stream>>>(
            agg, bna, bnc, batch, (layer == 0) ? h0 : nullptr, out,
            layer * HD, nnodes);
    }
}